// TemporalAttentionPooling_12008728559818
// MI455X (gfx1250) — compile-verified
//
#include <hip/hip_runtime.h>
#include <hip/hip_bf16.h>

#define B_  64
#define T_  1000
#define F_  512
#define H_  8
#define KD_ 32
#define HK_ 256
#define EPS_ 1e-6f

typedef float v2f __attribute__((ext_vector_type(2)));
typedef float v8f __attribute__((ext_vector_type(8)));

__device__ __forceinline__ float wave_sum(float v) {
#pragma unroll
  for (int off = 16; off; off >>= 1) v += __shfl_xor(v, off, 32);
  return v;
}
__device__ __forceinline__ float wave_max(float v) {
#pragma unroll
  for (int off = 16; off; off >>= 1) v = fmaxf(v, __shfl_xor(v, off, 32));
  return v;
}

// ---------------- K1: qh, wk_eff, c[h] ----------------
__global__ void k_prep(const float* __restrict__ q, const float* __restrict__ Wq,
                       const float* __restrict__ bq, const float* __restrict__ Wk,
                       const float* __restrict__ bk,
                       float* __restrict__ qh_out, float* __restrict__ wkeff,
                       float* __restrict__ ch) {
  __shared__ float qh_s[HK_];
  int tid = threadIdx.x;
  float acc = bq[tid];
  for (int f = 0; f < F_; ++f) acc += q[f] * Wq[f * HK_ + tid];
  qh_s[tid] = acc;
  qh_out[tid] = acc;
  __syncthreads();
  for (int idx = tid; idx < H_ * F_; idx += 256) {
    int h = idx >> 9, f = idx & 511;
    float s = 0.f;
#pragma unroll
    for (int k = 0; k < KD_; ++k) s += qh_s[h * KD_ + k] * Wk[f * HK_ + h * KD_ + k];
    wkeff[h * F_ + f] = s;
  }
  if (tid < H_) {
    float s = 0.f;
#pragma unroll
    for (int k = 0; k < KD_; ++k) s += qh_s[tid * KD_ + k] * bk[tid * KD_ + k];
    ch[tid] = s;
  }
}

// ---------------- K2: scores[b,h,t] = scale*(x_row . wk_eff[h] + c[h]) ----------------
// wave-per-row, 8 rows per wave, 8 waves per block -> 64 rows/block, 1000 blocks
__global__ void k_scores(const float* __restrict__ x, const float* __restrict__ wkeff,
                         const float* __restrict__ ch, float* __restrict__ scores) {
  __shared__ float wks[H_ * F_];
  __shared__ float chs[H_];
  int tid = threadIdx.x;
  for (int i = tid; i < H_ * F_; i += 256) wks[i] = wkeff[i];
  if (tid < H_) chs[tid] = ch[tid];
  __syncthreads();
  int lane = tid & 31, wave = tid >> 5;
  const float scale = 0.17677669529663687f;  // 1/sqrt(32)
  for (int i = 0; i < 8; ++i) {
    int r = blockIdx.x * 64 + wave * 8 + i;
    int b = r / T_, t = r - b * T_;
    const float* xr = x + ((size_t)b * T_ + t) * F_;
    float acc[H_];
#pragma unroll
    for (int h = 0; h < H_; ++h) acc[h] = 0.f;
#pragma unroll
    for (int j = 0; j < 4; ++j) {
      int f0 = (j * 32 + lane) * 4;
      float4 xv = *(const float4*)(xr + f0);
#pragma unroll
      for (int h = 0; h < H_; ++h) {
        float4 wv = *(const float4*)(&wks[h * F_ + f0]);
        acc[h] += xv.x * wv.x + xv.y * wv.y + xv.z * wv.z + xv.w * wv.w;
      }
    }
#pragma unroll
    for (int h = 0; h < H_; ++h) {
      float v = wave_sum(acc[h]);
      if (lane == 0) scores[((size_t)b * H_ + h) * T_ + t] = (v + chs[h]) * scale;
    }
  }
}

// ---------------- K3: softmax over T per (b,h) ----------------
__global__ void k_softmax(float* __restrict__ scores) {
  int bh = blockIdx.x, tid = threadIdx.x;
  int lane = tid & 31, wave = tid >> 5;
  float* s = scores + (size_t)bh * T_;
  __shared__ float r0[8], r1[8];
  float m = -3.402823466e+38f;
  for (int i = tid; i < T_; i += 256) m = fmaxf(m, s[i]);
  m = wave_max(m);
  if (lane == 0) r0[wave] = m;
  __syncthreads();
  if (tid == 0) {
    float mm = r0[0];
#pragma unroll
    for (int i = 1; i < 8; ++i) mm = fmaxf(mm, r0[i]);
    r0[0] = mm;
  }
  __syncthreads();
  float mx = r0[0];
  float sum = 0.f;
  for (int i = tid; i < T_; i += 256) {
    float e = expf(s[i] - mx);
    s[i] = e;
    sum += e;
  }
  sum = wave_sum(sum);
  if (lane == 0) r1[wave] = sum;
  __syncthreads();
  if (tid == 0) {
    float ss = 0.f;
#pragma unroll
    for (int i = 0; i < 8; ++i) ss += r1[i];
    r1[0] = ss;
  }
  __syncthreads();
  float inv = 1.0f / r1[0];
  for (int i = tid; i < T_; i += 256) s[i] *= inv;
}

// ---------------- K4a: partial xbar over a T-slice ----------------
// grid = (B, S); block s accumulates t in [s*TS, min((s+1)*TS, T))
#define CHUNK_ 128
__global__ void k_xbar_part(const float* __restrict__ x, const float* __restrict__ attn,
                            float* __restrict__ xpart, int TS) {
  int b = blockIdx.x, s = blockIdx.y, tid = threadIdx.x;
  int t0 = s * TS;
  int t1 = t0 + TS;
  if (t1 > T_) t1 = T_;
  __shared__ float as_[H_][CHUNK_];
  float acc0[H_], acc1[H_];
#pragma unroll
  for (int h = 0; h < H_; ++h) { acc0[h] = 0.f; acc1[h] = 0.f; }
  for (int c0 = t0; c0 < t1; c0 += CHUNK_) {
    int valid = (t1 - c0 < CHUNK_) ? (t1 - c0) : CHUNK_;
    __syncthreads();
    for (int idx = tid; idx < H_ * CHUNK_; idx += 256) {
      int h = idx / CHUNK_, i = idx - h * CHUNK_;
      as_[h][i] = (i < valid) ? attn[((size_t)b * H_ + h) * T_ + c0 + i] : 0.f;
    }
    __syncthreads();
    for (int i = 0; i < valid; ++i) {
      const float* xr = x + ((size_t)b * T_ + c0 + i) * F_;
      if ((i & 7) == 0) __builtin_prefetch(xr + 8 * F_ + tid, 0, 0);  // global_prefetch_b8
      float xv0 = xr[tid];
      float xv1 = xr[tid + 256];
#pragma unroll
      for (int h = 0; h < H_; ++h) {
        float a = as_[h][i];
        acc0[h] += a * xv0;
        acc1[h] += a * xv1;
      }
    }
  }
  size_t base = ((size_t)s * B_ + b) * (H_ * F_);
#pragma unroll
  for (int h = 0; h < H_; ++h) {
    xpart[base + h * F_ + tid] = acc0[h];
    xpart[base + h * F_ + tid + 256] = acc1[h];
  }
}

// ---------------- K4b: reduce S partial slices (deterministic) ----------------
__global__ void k_xbar_reduce(const float* __restrict__ xpart, float* __restrict__ xbar,
                              int S) {
  size_t i = (size_t)blockIdx.x * 256 + threadIdx.x;  // B*H*F total
  float s = 0.f;
  for (int k = 0; k < S; ++k) s += xpart[(size_t)k * (B_ * H_ * F_) + i];
  xbar[i] = s;
}

// ---------------- K5: ctx[b,hk] = xbar[b,h,:] . Wv[:,hk] + bv ----------------
__global__ void k_ctx(const float* __restrict__ xbar, const float* __restrict__ Wv,
                      const float* __restrict__ bv, float* __restrict__ ctx) {
  int b = blockIdx.x, tid = threadIdx.x;
  __shared__ float xs[H_ * F_];
  for (int i = tid; i < H_ * F_; i += 256) xs[i] = xbar[(size_t)b * H_ * F_ + i];
  __syncthreads();
  int h = tid >> 5;
  float s = bv[tid];
  const float* xr = xs + h * F_;
  for (int f = 0; f < F_; ++f) s += xr[f] * Wv[f * HK_ + tid];
  ctx[b * HK_ + tid] = s;
}

// ---------------- K6: pooled[b,f] = ctx[b,:] . Wo[:,f] + bo ----------------
__global__ void k_pooled(const float* __restrict__ ctx, const float* __restrict__ Wo,
                         const float* __restrict__ bo, float* __restrict__ pooled) {
  int b = blockIdx.x, tid = threadIdx.x;
  __shared__ float cs[HK_];
  cs[tid] = ctx[b * HK_ + tid];
  __syncthreads();
  float s0 = bo[tid], s1 = bo[tid + 256];
  for (int hk = 0; hk < HK_; ++hk) {
    float c = cs[hk];
    s0 += c * Wo[hk * F_ + tid];
    s1 += c * Wo[hk * F_ + tid + 256];
  }
  pooled[b * F_ + tid] = s0;
  pooled[b * F_ + tid + 256] = s1;
}

// ---------------- K7: preln = pooled(64x512) @ Wd(512x512), fp32 WMMA ----------------
// one wave per 16x16 output tile; 4 M-tiles x 32 N-tiles = 128 waves = 16 blocks x 8 waves
__global__ void k_gemm_wmma(const float* __restrict__ pooled, const float* __restrict__ Wd,
                            float* __restrict__ preln) {
  int tid = threadIdx.x;
  int lane = tid & 31, wave = tid >> 5;
  int w = blockIdx.x * 8 + wave;
  int mt = w >> 5;       // 0..3
  int nt = w & 31;       // 0..31
  int half = lane >> 4;  // lane halves carry K pairs
  int l16 = lane & 15;
  v8f c = {0.f, 0.f, 0.f, 0.f, 0.f, 0.f, 0.f, 0.f};
#pragma unroll 4
  for (int k0 = 0; k0 < F_; k0 += 4) {
    // A fragment: 16x4, rows M=l16, K = k0 + half*2 + {0,1}
    v2f a;
    const float* ap = pooled + (mt * 16 + l16) * F_ + k0 + half * 2;
    a.x = ap[0];
    a.y = ap[1];
    // B fragment: 4x16, cols N=l16, K = k0 + half*2 + {0,1} (row-major Wd)
    v2f bb;
    bb.x = Wd[(k0 + half * 2) * F_ + nt * 16 + l16];
    bb.y = Wd[(k0 + half * 2 + 1) * F_ + nt * 16 + l16];
    c = __builtin_amdgcn_wmma_f32_16x16x4_f32(false, a, false, bb, (short)0, c,
                                              false, false);
  }
  // C/D layout: VGPR r -> lanes0-15: M=r, lanes16-31: M=r+8; N=lane&15
#pragma unroll
  for (int r = 0; r < 8; ++r) {
    int m = mt * 16 + r + half * 8;
    int n = nt * 16 + l16;
    preln[m * F_ + n] = c[r];
  }
}

// ---------------- K8: LayerNorm ----------------
__global__ void k_ln(const float* __restrict__ preln, const float* __restrict__ gamma,
                     const float* __restrict__ beta, float* __restrict__ out) {
  int b = blockIdx.x, tid = threadIdx.x;
  int lane = tid & 31, wave = tid >> 5;
  __shared__ float rs[8], rq[8];
  float v0 = preln[b * F_ + tid];
  float v1 = preln[b * F_ + tid + 256];
  float s = v0 + v1;
  float q = v0 * v0 + v1 * v1;
  s = wave_sum(s);
  q = wave_sum(q);
  if (lane == 0) { rs[wave] = s; rq[wave] = q; }
  __syncthreads();
  if (tid == 0) {
    float S = 0.f, Q = 0.f;
#pragma unroll
    for (int i = 0; i < 8; ++i) { S += rs[i]; Q += rq[i]; }
    rs[0] = S;
    rq[0] = Q;
  }
  __syncthreads();
  float mean = rs[0] * (1.0f / F_);
  float var = rq[0] * (1.0f / F_) - mean * mean;
  float inv = rsqrtf(var + EPS_);
  out[b * F_ + tid] = (v0 - mean) * inv * gamma[tid] + beta[tid];
  out[b * F_ + tid + 256] = (v1 - mean) * inv * gamma[tid + 256] + beta[tid + 256];
}

extern "C" void kernel_launch(void* const* d_in, const int* in_sizes, int n_in,
                              void* d_out, int out_size, void* d_ws, size_t ws_size,
                              hipStream_t stream) {
  const float* x     = (const float*)d_in[0];
  const float* q     = (const float*)d_in[1];
  const float* Wq    = (const float*)d_in[2];
  const float* bq    = (const float*)d_in[3];
  const float* Wk    = (const float*)d_in[4];
  const float* bk    = (const float*)d_in[5];
  const float* Wv    = (const float*)d_in[6];
  const float* bv    = (const float*)d_in[7];
  const float* Wo    = (const float*)d_in[8];
  const float* bo    = (const float*)d_in[9];
  const float* Wd    = (const float*)d_in[10];
  const float* gamma = (const float*)d_in[11];
  const float* beta  = (const float*)d_in[12];
  float* out = (float*)d_out;

  float* w = (float*)d_ws;
  float* qh     = w;                                // 256
  float* ch     = w + 256;                          // 16 (padded)
  float* wkeff  = w + 272;                          // 4096
  float* scores = w + 4368;                         // B*H*T = 512000
  float* xbar   = scores + (size_t)B_ * H_ * T_;    // 262144
  float* ctx    = xbar + (size_t)B_ * H_ * F_;      // 16384
  float* pooled = ctx + (size_t)B_ * HK_;           // 32768
  float* preln  = pooled + (size_t)B_ * F_;         // 32768
  float* xpart  = preln + (size_t)B_ * F_;          // S * 262144 (if it fits)

  const size_t base_floats = 4368 + (size_t)B_ * H_ * T_ + (size_t)B_ * H_ * F_ +
                             (size_t)B_ * HK_ + 2 * (size_t)B_ * F_;
  // pick largest slice count S (for pass-2 occupancy) that fits in workspace;
  // ws_size is constant per run -> deterministic choice
  int S = 16;
  while (S > 1 &&
         (base_floats + (size_t)S * (B_ * H_ * F_)) * sizeof(float) > ws_size)
    S >>= 1;
  bool have_part =
      (base_floats + (size_t)S * (B_ * H_ * F_)) * sizeof(float) <= ws_size;
  if (!have_part) { S = 1; xpart = xbar; }  // in-place fallback, no reduce
  int TS = (T_ + S - 1) / S;

  k_prep<<<1, 256, 0, stream>>>(q, Wq, bq, Wk, bk, qh, wkeff, ch);
  k_scores<<<(B_ * T_) / 64, 256, 0, stream>>>(x, wkeff, ch, scores);
  k_softmax<<<B_ * H_, 256, 0, stream>>>(scores);
  k_xbar_part<<<dim3(B_, S), 256, 0, stream>>>(x, scores, xpart, TS);
  if (xpart != xbar)
    k_xbar_reduce<<<(B_ * H_ * F_) / 256, 256, 0, stream>>>(xpart, xbar, S);
  k_ctx<<<B_, 256, 0, stream>>>(xbar, Wv, bv, ctx);
  k_pooled<<<B_, 256, 0, stream>>>(ctx, Wo, bo, pooled);
  k_gemm_wmma<<<16, 256, 0, stream>>>(pooled, Wd, preln);
  k_ln<<<B_, 256, 0, stream>>>(preln, gamma, beta, out);
}